// InferenceUpdater_22643067585019
// MI455X (gfx1250) — compile-verified
//
#include <hip/hip_runtime.h>
#include <hip/hip_bf16.h>

typedef __attribute__((ext_vector_type(2))) float v2f;
typedef __attribute__((ext_vector_type(8))) float v8f;
typedef __attribute__((ext_vector_type(4))) int   v4i;

#if __has_builtin(__builtin_amdgcn_global_load_async_to_lds_b128) && \
    __has_builtin(__builtin_amdgcn_s_wait_asynccnt)
#define USE_ASYNC 1
#else
#define USE_ASYNC 0
#endif

// Problem constants
constexpr int Bn   = 4;
constexpr int Nn   = 4;
constexpr int Cc   = 64;
constexpr int HWn  = 160 * 160;          // 25600
constexpr int VS   = 132;                // LDS row stride for [channel][pixel] tiles
                                         // (132*4 B = 528 B: 16B-aligned rows, banks (4c+p)%64)
// d_out layout (f32 elements)
constexpr size_t OFF_STATE   = 26214400;  // B*N*C*HW
constexpr size_t OFF_KEYSUM  = 26279936;  // + B*N*C*C
constexpr size_t OFF_OBJ     = 32833536;  // + B*C*HW

// ---------------------------------------------------------------------------
// Kernel 0: new_state = gate*state ; new_obj_mem_sum ; zero norm scratch
// ---------------------------------------------------------------------------
__global__ void iu_init_kernel(const float* __restrict__ state,
                               const float* __restrict__ gate,
                               const float* __restrict__ obj_mem,
                               const float* __restrict__ obj_mem_sum,
                               float* __restrict__ out_state,
                               float* __restrict__ out_obj,
                               float* __restrict__ ws_norm) {
  int idx = blockIdx.x * 256 + threadIdx.x;
  if (idx < Bn) ws_norm[idx] = 0.0f;
  if (idx < Bn * Nn * Cc * Cc) {
    int b = idx >> 14;            // N*C*C = 16384 per batch
    int v = idx & (Cc - 1);       // last axis (broadcast axis of gate)
    out_state[idx] = state[idx] * gate[b * Cc + v];
  }
  if (idx < Bn * Nn * 16 * Cc) {
    out_obj[idx] = obj_mem[idx] + obj_mem_sum[idx];
  }
}

// ---------------------------------------------------------------------------
// Async helper: one wave issues 16 b128 async copies covering its 16 channel
// rows of a 64(ch) x 128(px) value tile.  Global side: 32 lanes x 16B = one
// full 512B row per instruction (perfectly coalesced).  LDS side lands in
// [channel][pixel] stride-132 layout.
// ---------------------------------------------------------------------------
#if USE_ASYNC
__device__ __forceinline__ void issue_value_tile(const float* __restrict__ vtile,
                                                 float* __restrict__ buf,
                                                 int w, int lane) {
#pragma unroll
  for (int i = 0; i < 16; ++i) {
    const int c = i * 4 + w;
    const float* g = vtile + (size_t)c * HWn + lane * 4;
    float* l = buf + c * VS + lane * 4;
    __builtin_amdgcn_global_load_async_to_lds_b128(
        (__attribute__((address_space(1))) v4i*)g,
        (__attribute__((address_space(3))) v4i*)l,
        0, 0);
  }
}
#endif

// ---------------------------------------------------------------------------
// Kernel 1: softmax + new_key_sum + norm partials + state-write GEMM
//   block = 128 threads (4 waves), handles (b, 128-pixel chunk)
// ---------------------------------------------------------------------------
__launch_bounds__(128)
__global__ void iu_pass1_kernel(const float* __restrict__ key_in,
                                const float* __restrict__ key_sum_in,
                                const float* __restrict__ value_in,
                                float* __restrict__ out_key_sum,
                                float* __restrict__ out_state,
                                float* __restrict__ ws_key,
                                float* __restrict__ ws_norm) {
  __shared__ __align__(16) float lds_key[Cc * VS];       // [c][pixel]
  __shared__ __align__(16) float lds_val[2][Cc * VS];    // [c][pixel], double buffer

  const int t   = threadIdx.x;
  const int b   = blockIdx.y;
  const int hw0 = blockIdx.x * 128;
  const int p   = hw0 + t;
  const int w    = t >> 5;        // wave id
  const int lane = t & 31;
  const int col  = lane & 15;
  const int half = lane >> 4;     // 0: lanes 0-15, 1: lanes 16-31

  const float* vbase = value_in + ((size_t)b * Nn) * Cc * HWn + hw0;

#if USE_ASYNC
  // Kick off value tile n=0 immediately: overlaps with the whole softmax phase.
  issue_value_tile(vbase, &lds_val[0][0], w, lane);
#endif

  // ---- Phase A: channel softmax, thread-per-pixel (coalesced per channel) --
  const float* kp = key_in + (size_t)b * Cc * HWn + p;
  float kv[Cc];
  float mx = -3.4e38f;
#pragma unroll
  for (int c = 0; c < Cc; ++c) { kv[c] = kp[(size_t)c * HWn]; mx = fmaxf(mx, kv[c]); }
  float sum = 0.0f;
#pragma unroll
  for (int c = 0; c < Cc; ++c) {
    kv[c] = exp2f((kv[c] - mx) * 1.4426950408889634f);
    sum += kv[c];
  }
  const float rs = 1.0f / sum;

  const float* ksp = key_sum_in + (size_t)b * Cc * HWn + p;
  float* oks = out_key_sum + (size_t)b * Cc * HWn + p;
  float* wk  = ws_key      + (size_t)b * Cc * HWn + p;
  float np = 0.0f;
#pragma unroll
  for (int c = 0; c < Cc; ++c) {
    float kc = kv[c] * rs;
    float ks = ksp[(size_t)c * HWn];
    np = fmaf(kc, ks, np);
    oks[(size_t)c * HWn] = ks + kc;   // new_key_sum
    wk [(size_t)c * HWn] = kc;        // softmaxed key for pass 2
    lds_key[c * VS + t] = kc;         // transposed stash for A fragments
  }
  // norm partial: wave reduce, one atomic per wave
#pragma unroll
  for (int off = 16; off > 0; off >>= 1) np += __shfl_down(np, off, 32);
  if (lane == 0) atomicAdd(&ws_norm[b], np);
  __syncthreads();    // lds_key ready for all waves

  // ---- Phase B: new_state += key @ value^T over this pixel chunk ----------
  for (int n = 0; n < Nn; ++n) {
    float* vbuf = &lds_val[n & 1][0];
#if USE_ASYNC
    if (n > 0) __syncthreads();            // buf[(n+1)&1] free to overwrite
    if (n < Nn - 1)
      issue_value_tile(vbase + (size_t)(n + 1) * Cc * HWn,
                       &lds_val[(n + 1) & 1][0], w, lane);
    if (n < Nn - 1) __builtin_amdgcn_s_wait_asynccnt(16);  // tile n landed
    else            __builtin_amdgcn_s_wait_asynccnt(0);
    __syncthreads();                       // all waves' tile-n copies visible
#else
    if (n > 0) __syncthreads();
    const float* vp = vbase + (size_t)n * Cc * HWn + t;
#pragma unroll
    for (int c = 0; c < Cc; ++c) vbuf[c * VS + t] = vp[(size_t)c * HWn];
    __syncthreads();
#endif

#pragma unroll
    for (int vt = 0; vt < 4; ++vt) {
      v8f acc = {};
#pragma unroll 4
      for (int s4 = 0; s4 < 32; ++s4) {        // 128 pixels / K=4
        const int pb = s4 * 4 + half * 2;
        // A: key[k-ch, hw..hw+1]  (b64, pixels consecutive)
        v2f a  = *(const v2f*)&lds_key[(w  * 16 + col) * VS + pb];
        // B: value^T[hw..hw+1, v] (b64, pixels consecutive)
        v2f bf = *(const v2f*)&vbuf   [(vt * 16 + col) * VS + pb];
        acc = __builtin_amdgcn_wmma_f32_16x16x4_f32(
            false, a, false, bf, (short)0, acc, false, false);
      }
      // D 16x16: VGPR r -> row r (lanes 0-15) / r+8 (lanes 16-31)
      float* os = out_state +
          (((size_t)b * Nn + n) * Cc + w * 16 + half * 8) * Cc + vt * 16 + col;
#pragma unroll
      for (int r = 0; r < 8; ++r) atomicAdd(os + (size_t)r * Cc, acc[r]);
    }
  }
}

// ---------------------------------------------------------------------------
// Kernel 2: readout = (state^T @ key) / norm[b]
//   block = 128 threads (4 waves), handles (64-pixel chunk, b, n)
// ---------------------------------------------------------------------------
__launch_bounds__(128)
__global__ void iu_pass2_kernel(const float* __restrict__ state_in,
                                const float* __restrict__ ws_key,
                                const float* __restrict__ ws_norm,
                                float* __restrict__ out_readout) {
  __shared__ float lds_state[Cc * 66];   // [v][k], pad 66 (b64-aligned)
  __shared__ float lds_key2[64 * 66];    // [pixel][k-channel], pad 66

  const int t   = threadIdx.x;
  const int b   = blockIdx.y;
  const int n   = blockIdx.z;
  const int hw0 = blockIdx.x * 64;

  // stage state[b,n] transposed: lds_state[v*66 + k] = state[b,n,k,v]
  const float* sp = state_in + ((size_t)b * Nn + n) * Cc * Cc;
#pragma unroll
  for (int i = 0; i < Cc * Cc; i += 128) {
    int idx = i + t;
    int k = idx >> 6, v = idx & 63;
    lds_state[v * 66 + k] = sp[idx];
  }
  // stage key tile: lds_key2[p*66 + c] = key[b, c, hw0+p]
  const float* kb = ws_key + (size_t)b * Cc * HWn + hw0;
#pragma unroll
  for (int i = 0; i < Cc * 64; i += 128) {
    int idx = i + t;
    int c = idx >> 6, p = idx & 63;
    lds_key2[p * 66 + c] = kb[(size_t)c * HWn + p];
  }
  __syncthreads();

  const float rnorm = 1.0f / ws_norm[b];
  const int w    = t >> 5;        // wave id -> v-tile
  const int lane = t & 31;
  const int col  = lane & 15;
  const int half = lane >> 4;

#pragma unroll
  for (int ht = 0; ht < 4; ++ht) {         // 4 hw sub-tiles of 16 pixels
    v8f acc = {};
#pragma unroll
    for (int kk = 0; kk < 16; ++kk) {      // K = 64 in steps of 4
      // A: state^T[v, k..k+1]  (b64 load, k consecutive)
      v2f a  = *(const v2f*)&lds_state[(w  * 16 + col) * 66 + kk * 4 + half * 2];
      // B: key[k..k+1, pixel]  (b64 load, channel consecutive)
      v2f bf = *(const v2f*)&lds_key2 [(ht * 16 + col) * 66 + kk * 4 + half * 2];
      acc = __builtin_amdgcn_wmma_f32_16x16x4_f32(
          false, a, false, bf, (short)0, acc, false, false);
    }
    float* o = out_readout +
        (((size_t)b * Nn + n) * Cc + w * 16 + half * 8) * HWn + hw0 + ht * 16 + col;
#pragma unroll
    for (int r = 0; r < 8; ++r) o[(size_t)r * HWn] = acc[r] * rnorm;
  }
}

// ---------------------------------------------------------------------------
extern "C" void kernel_launch(void* const* d_in, const int* in_sizes, int n_in,
                              void* d_out, int out_size, void* d_ws, size_t ws_size,
                              hipStream_t stream) {
  const float* key_in     = (const float*)d_in[0];
  const float* state_in   = (const float*)d_in[1];
  const float* key_sum_in = (const float*)d_in[2];
  const float* gate_in    = (const float*)d_in[3];
  const float* value_in   = (const float*)d_in[4];
  const float* obj_mem    = (const float*)d_in[5];
  const float* obj_sum    = (const float*)d_in[6];

  float* out         = (float*)d_out;
  float* out_readout = out;
  float* out_state   = out + OFF_STATE;
  float* out_keysum  = out + OFF_KEYSUM;
  float* out_obj     = out + OFF_OBJ;

  float* ws_norm = (float*)d_ws;                      // 4 floats
  float* ws_key  = (float*)((char*)d_ws + 256);       // B*C*HW floats

  iu_init_kernel<<<dim3(256), dim3(256), 0, stream>>>(
      state_in, gate_in, obj_mem, obj_sum, out_state, out_obj, ws_norm);

  iu_pass1_kernel<<<dim3(HWn / 128, Bn), dim3(128), 0, stream>>>(
      key_in, key_sum_in, value_in, out_keysum, out_state, ws_key, ws_norm);

  iu_pass2_kernel<<<dim3(HWn / 64, Bn, Nn), dim3(128), 0, stream>>>(
      state_in, ws_key, ws_norm, out_readout);
}